// clDiceLoss_16329465659785
// MI455X (gfx1250) — compile-verified
//
#include <hip/hip_runtime.h>
#include <hip/hip_bf16.h>
#include <stdint.h>

// clDice loss on MI455X (gfx1250).
// pred/target = (32, 1, 512, 512) f32; output = scalar f32.

#define HDIM 512
#define WDIM 512
#define HW   (HDIM * WDIM)

#define TX   32             // outputs per block in x
#define TY   8              // outputs per block in y
#define NTHR (TX * TY)      // 256 threads = 8 waves
#define PITCH 40            // LDS row pitch (floats)

typedef float v2f __attribute__((ext_vector_type(2)));
typedef float v8f __attribute__((ext_vector_type(8)));

// ---------------------------------------------------------------- sigmoid ---
__global__ void k_sigmoid(const float* __restrict__ in, float* __restrict__ out, int n) {
    int idx = blockIdx.x * blockDim.x + threadIdx.x;
    if (idx >= n) return;
    __builtin_prefetch(in + idx + 1024, 0, 0);      // gfx1250 global_prefetch_b8
    out[idx] = 1.0f / (1.0f + expf(-in[idx]));
}

// ------------------------------------------ async region load into LDS ------
// Loads an (lx x ly) region whose top-left is (x0, y0) (may be negative;
// coordinates clamped so loads are always valid — borders handled by guards
// at use).  gfx1250 GLOBAL_LOAD_ASYNC_TO_LDS_B32, tracked by ASYNCcnt.
__device__ __forceinline__ void async_load_region(const float* __restrict__ img,
                                                  int x0, int y0, int lx, int ly,
                                                  float* lds) {
    for (int i = threadIdx.x; i < lx * ly; i += NTHR) {
        int r = i / lx;
        int c = i - r * lx;
        int gy = min(max(y0 + r, 0), HDIM - 1);
        int gx = min(max(x0 + c, 0), WDIM - 1);
        const float* gp = img + gy * WDIM + gx;
        unsigned ldsoff = (unsigned)(unsigned long long)(const void*)(lds + r * PITCH + c);
        asm volatile("global_load_async_to_lds_b32 %0, %1, off"
                     :: "v"(ldsoff), "v"(gp) : "memory");
    }
    asm volatile("s_wait_asynccnt 0x0" ::: "memory");
    __syncthreads();
}

// Plus-shaped 5-point min (soft_erode) with global-bounds guards.
__device__ __forceinline__ float erode_at(const float* t, int r, int c, int gx, int gy) {
    // t indexed so that (r, c) is the center element for global (gx, gy)
    float ctr = t[r * PITCH + c];
    float mv = ctr, mh = ctr;
    if (gy > 0)        mv = fminf(mv, t[(r - 1) * PITCH + c]);
    if (gy < HDIM - 1) mv = fminf(mv, t[(r + 1) * PITCH + c]);
    if (gx > 0)        mh = fminf(mh, t[r * PITCH + c - 1]);
    if (gx < WDIM - 1) mh = fminf(mh, t[r * PITCH + c + 1]);
    return fminf(mv, mh);
}

// 3x3 max (soft_dilate) with global-bounds guards.
__device__ __forceinline__ float dilate_at(const float* t, int r, int c, int gx, int gy) {
    bool up = gy > 0, dn = gy < HDIM - 1, lf = gx > 0, rt = gx < WDIM - 1;
    float m = t[r * PITCH + c];
    if (up)       m = fmaxf(m, t[(r - 1) * PITCH + c]);
    if (dn)       m = fmaxf(m, t[(r + 1) * PITCH + c]);
    if (lf)       m = fmaxf(m, t[r * PITCH + c - 1]);
    if (rt)       m = fmaxf(m, t[r * PITCH + c + 1]);
    if (up && lf) m = fmaxf(m, t[(r - 1) * PITCH + c - 1]);
    if (up && rt) m = fmaxf(m, t[(r - 1) * PITCH + c + 1]);
    if (dn && lf) m = fmaxf(m, t[(r + 1) * PITCH + c - 1]);
    if (dn && rt) m = fmaxf(m, t[(r + 1) * PITCH + c + 1]);
    return m;
}

// ----------------------------------------- fused init: relu(img - open) -----
// skel = relu(img - dilate(erode(img))).  Halo-2 tile; erode staged in LDS.
__global__ void k_init_fused(const float* __restrict__ img, float* __restrict__ skel) {
    __shared__ float t0[(TY + 4) * PITCH];   // img tile: 12 x 36 region
    __shared__ float e1[(TY + 2) * PITCH];   // erode:    10 x 34 region
    int x0 = blockIdx.x * TX, y0 = blockIdx.y * TY, b = blockIdx.z;
    const float* im = img + (size_t)b * HW;

    async_load_region(im, x0 - 2, y0 - 2, TX + 4, TY + 4, t0);

    // e1[r][c] = erode(img) at global (y0-1+r, x0-1+c), region 10 x 34
    for (int i = threadIdx.x; i < (TY + 2) * (TX + 2); i += NTHR) {
        int r = i / (TX + 2);
        int c = i - r * (TX + 2);
        e1[r * PITCH + c] = erode_at(t0, r + 1, c + 1, x0 - 1 + c, y0 - 1 + r);
    }
    __syncthreads();

    int tx = threadIdx.x & (TX - 1);
    int ty = threadIdx.x >> 5;
    int gx = x0 + tx, gy = y0 + ty;
    float o = dilate_at(e1, ty + 1, tx + 1, gx, gy);
    float c0 = t0[(ty + 2) * PITCH + (tx + 2)];          // img center from tile
    skel[(size_t)b * HW + (size_t)gy * WDIM + gx] = fmaxf(c0 - o, 0.0f);
}

// --------------------- fused iteration: one kernel per skeleton iteration ---
// e1 = erode(cur); e2 = erode(e1); o = dilate(e2);
// delta = relu(e1 - o); skel += relu(delta - skel*delta); xn = e1.
// Radius-3 dependency: halo-3 tile, all intermediates staged in LDS.
__global__ void k_iter_fused(const float* __restrict__ cur, float* __restrict__ xn,
                             float* __restrict__ skel) {
    __shared__ float t0[(TY + 6) * PITCH];   // cur tile: 14 x 38 region
    __shared__ float e1[(TY + 4) * PITCH];   // erode^1:  12 x 36 region
    __shared__ float e2[(TY + 2) * PITCH];   // erode^2:  10 x 34 region
    int x0 = blockIdx.x * TX, y0 = blockIdx.y * TY, b = blockIdx.z;
    const float* im = cur + (size_t)b * HW;

    async_load_region(im, x0 - 3, y0 - 3, TX + 6, TY + 6, t0);

    // e1[r][c] at global (y0-2+r, x0-2+c), region 12 x 36
    for (int i = threadIdx.x; i < (TY + 4) * (TX + 4); i += NTHR) {
        int r = i / (TX + 4);
        int c = i - r * (TX + 4);
        e1[r * PITCH + c] = erode_at(t0, r + 1, c + 1, x0 - 2 + c, y0 - 2 + r);
    }
    __syncthreads();

    // e2[r][c] at global (y0-1+r, x0-1+c), region 10 x 34
    for (int i = threadIdx.x; i < (TY + 2) * (TX + 2); i += NTHR) {
        int r = i / (TX + 2);
        int c = i - r * (TX + 2);
        e2[r * PITCH + c] = erode_at(e1, r + 1, c + 1, x0 - 1 + c, y0 - 1 + r);
    }
    __syncthreads();

    int tx = threadIdx.x & (TX - 1);
    int ty = threadIdx.x >> 5;
    int gx = x0 + tx, gy = y0 + ty;
    size_t idx = (size_t)b * HW + (size_t)gy * WDIM + gx;

    float xv = e1[(ty + 2) * PITCH + (tx + 2)];          // x_new = erode(cur)
    float o  = dilate_at(e2, ty + 1, tx + 1, gx, gy);    // open(x_new)
    float delta = fmaxf(xv - o, 0.0f);
    float s = skel[idx];
    skel[idx] = s + fmaxf(delta - s * delta, 0.0f);
    xn[idx] = xv;
}

// --------------------------------------------------- stage-1 WMMA reduce ----
// Four sums at once. A = ones; V_WMMA_F32_16X16X4_F32 folds 64-float chunks
// into K exactly (f32).  sum(c[0] over 32 lanes) = 2 * total.
__global__ void k_reduce_stage1(const float* __restrict__ skp, const float* __restrict__ tgt,
                                const float* __restrict__ skt, const float* __restrict__ prob,
                                float* __restrict__ partials, int nChunks) {
    int lane = threadIdx.x & 31;
    int wave = threadIdx.x >> 5;
    int wavesPerBlock = blockDim.x >> 5;
    int gw = blockIdx.x * wavesPerBlock + wave;
    int tw = gridDim.x * wavesPerBlock;

    v2f ones = {1.0f, 1.0f};
    v8f c1 = {}, c2 = {}, c3 = {}, c4 = {};

    for (int ch = gw; ch < nChunks; ch += tw) {          // wave-uniform: EXEC all ones
        size_t base = (size_t)ch * 64 + (size_t)(lane * 2);
        v2f a = *(const v2f*)(skp  + base);
        v2f t = *(const v2f*)(tgt  + base);
        v2f s = *(const v2f*)(skt  + base);
        v2f q = *(const v2f*)(prob + base);
        v2f b1 = a * t;
        v2f b2 = a;
        v2f b3 = s * q;
        v2f b4 = s;
        c1 = __builtin_amdgcn_wmma_f32_16x16x4_f32(false, ones, false, b1, (short)0, c1, false, false);
        c2 = __builtin_amdgcn_wmma_f32_16x16x4_f32(false, ones, false, b2, (short)0, c2, false, false);
        c3 = __builtin_amdgcn_wmma_f32_16x16x4_f32(false, ones, false, b3, (short)0, c3, false, false);
        c4 = __builtin_amdgcn_wmma_f32_16x16x4_f32(false, ones, false, b4, (short)0, c4, false, false);
    }

    float v1 = c1[0], v2 = c2[0], v3 = c3[0], v4 = c4[0];
    for (int m = 16; m > 0; m >>= 1) {                   // wave32 butterfly
        v1 += __shfl_xor(v1, m, 32);
        v2 += __shfl_xor(v2, m, 32);
        v3 += __shfl_xor(v3, m, 32);
        v4 += __shfl_xor(v4, m, 32);
    }

    __shared__ float sm[8][4];
    if (lane == 0) {
        sm[wave][0] = 0.5f * v1;   // rows 0 and 8 both carry full column sums
        sm[wave][1] = 0.5f * v2;
        sm[wave][2] = 0.5f * v3;
        sm[wave][3] = 0.5f * v4;
    }
    __syncthreads();
    if (threadIdx.x == 0) {
        float s0 = 0.f, s1 = 0.f, s2 = 0.f, s3 = 0.f;
        for (int w = 0; w < wavesPerBlock; ++w) {
            s0 += sm[w][0]; s1 += sm[w][1]; s2 += sm[w][2]; s3 += sm[w][3];
        }
        partials[blockIdx.x * 4 + 0] = s0;
        partials[blockIdx.x * 4 + 1] = s1;
        partials[blockIdx.x * 4 + 2] = s2;
        partials[blockIdx.x * 4 + 3] = s3;
    }
}

// -------------------------------------------------------- stage-2 + loss ----
__global__ void k_reduce_stage2(const float* __restrict__ partials, int nPart,
                                float* __restrict__ out) {
    __shared__ float sm[256][4];
    float s[4] = {0.f, 0.f, 0.f, 0.f};
    for (int i = threadIdx.x; i < nPart; i += blockDim.x)
        for (int k = 0; k < 4; ++k) s[k] += partials[i * 4 + k];
    for (int k = 0; k < 4; ++k) sm[threadIdx.x][k] = s[k];
    __syncthreads();
    for (int off = 128; off > 0; off >>= 1) {
        if ((int)threadIdx.x < off)
            for (int k = 0; k < 4; ++k) sm[threadIdx.x][k] += sm[threadIdx.x + off][k];
        __syncthreads();
    }
    if (threadIdx.x == 0) {
        float s1 = sm[0][0];   // sum(skel_pred * target)
        float s2 = sm[0][1];   // sum(skel_pred)
        float s3 = sm[0][2];   // sum(skel_target * pred_prob)
        float s4 = sm[0][3];   // sum(skel_target)
        float tprec = (s1 + 1.0f) / (s2 + 1.0f);
        float tsens = (s3 + 1.0f) / (s4 + 1.0f);
        float cl = 2.0f * tprec * tsens / (tprec + tsens + 1e-7f);
        out[0] = 1.0f - cl;
    }
}

// ---------------------------------------------------------- host sequence ---
static void run_skeleton(hipStream_t stream, const float* img, float* skel,
                         float* bufA, float* bufB) {
    dim3 grid(WDIM / TX, HDIM / TY, 32);
    dim3 block(NTHR);
    k_init_fused<<<grid, block, 0, stream>>>(img, skel);
    const float* cur = img;                 // never written
    float* pp[2] = {bufA, bufB};
    for (int it = 0; it < 10; ++it) {
        float* xn = pp[it & 1];
        k_iter_fused<<<grid, block, 0, stream>>>(cur, xn, skel);
        cur = xn;
    }
}

extern "C" void kernel_launch(void* const* d_in, const int* in_sizes, int n_in,
                              void* d_out, int out_size, void* d_ws, size_t ws_size,
                              hipStream_t stream) {
    const float* pred = (const float*)d_in[0];
    const float* tgt  = (const float*)d_in[1];
    float* out = (float*)d_out;
    const int n = in_sizes[0];           // 32*1*512*512 = 8388608

    float* wsf = (float*)d_ws;
    float* P        = wsf + 0ll * n;     // sigmoid(pred)
    float* SKP      = wsf + 1ll * n;     // skel(pred_prob)
    float* SKT      = wsf + 2ll * n;     // skel(target)
    float* X        = wsf + 3ll * n;     // ping
    float* T1       = wsf + 4ll * n;     // pong
    float* partials = wsf + 5ll * n;     // 1024 * 4 floats

    k_sigmoid<<<(n + 255) / 256, 256, 0, stream>>>(pred, P, n);

    run_skeleton(stream, P,   SKP, X, T1);
    run_skeleton(stream, tgt, SKT, X, T1);

    const int RB = 1024;                 // stage-1 blocks
    k_reduce_stage1<<<RB, 256, 0, stream>>>(SKP, tgt, SKT, P, partials, n / 64);
    k_reduce_stage2<<<1, 256, 0, stream>>>(partials, RB, out);
}